// Attention_22419729285290
// MI455X (gfx1250) — compile-verified
//
#include <hip/hip_runtime.h>
#include <hip/hip_bf16.h>
#include <math.h>

// ---------------------------------------------------------------------------
// Channel attention (Restormer-style) for MI455X / gfx1250.
//   DIM=96, HEADS=6 (16 ch/head), spatial N = 64^3 = 262144.
// fp32 end-to-end: V_WMMA_F32_16X16X4_F32 matrix pipe, B operands staged
// through LDS with GLOBAL_LOAD_ASYNC_TO_LDS_B128 (ASYNCcnt path).
// ---------------------------------------------------------------------------

typedef __attribute__((ext_vector_type(2))) float v2f;
typedef __attribute__((ext_vector_type(8))) float v8f;

#define NTOT (64LL * 64LL * 64LL)   // 262144
#define CDIM 96
#define QKV_C 288
#define HEADS 6

// --- gfx1250 async-copy helpers (CDNA5 ISA) -------------------
__device__ __forceinline__ unsigned lds_off32(const void* p) {
    // low 32 bits of a flat LDS pointer == byte offset within the workgroup's
    // LDS allocation (ISA 10.2 aperture calc), which is what the async-load
    // VDST operand wants.
    return (unsigned)(unsigned long long)p;
}
__device__ __forceinline__ void async_ld_b128(unsigned ldsoff, const float* gaddr) {
    asm volatile("global_load_async_to_lds_b128 %0, %1, off"
                 :: "v"(ldsoff), "v"(gaddr) : "memory");
}
__device__ __forceinline__ void wait_async0() {
    asm volatile("s_wait_asynccnt 0x0" ::: "memory");
}

// ---------------------------------------------------------------------------
// Kernels 1/5: Y[O x N] = W[O x 96] * X[96 x N] via WMMA f32 16x16x4.
//
// CDNA5 VGPR layouts (CDNA5 ISA):
//   A (16x4, MxK):  lane<16 -> M=lane, {v0,v1}={K0,K1}; lane>=16 -> {K2,K3}
//   B (4x16, KxN):  lane<16 -> N=lane, {v0,v1}={K0,K1}; lane>=16 -> {K2,K3}
//   C/D (16x16):    vgpr i, lane<16 -> (M=i, N=lane); lane>=16 -> (M=i+8, .)
//
// Block = 128 threads (4 waves) -> 16(M) x 64(N) tile. The 96x64 B-panel is
// async-copied to LDS once (row stride 80 floats: conflict-free ds reads),
// then 24 WMMAs consume it with immediate ds offsets.
// ---------------------------------------------------------------------------
#define BSTRIDE 80   // 64 data + 16 pad floats per staged row

__global__ __launch_bounds__(128) void gemm_wmma_f32(
    const float* __restrict__ W, const float* __restrict__ X,
    float* __restrict__ Y, long long N)
{
    __shared__ float tile[CDIM * BSTRIDE];        // 30 KB B-panel

    const int tid  = threadIdx.x;
    const int lane = tid & 31;
    const int wave = tid >> 5;
    const int r    = lane & 15;           // M (A) / N (B,D) sub-index
    const int hi   = lane >> 4;           // 0 -> K0,K1 ; 1 -> K2,K3
    const int kk   = hi << 1;

    const int       rowBase = blockIdx.y * 16;
    const long long col0    = (long long)blockIdx.x * 64;

    // ---- stage B panel: X[0..95][col0..col0+63] -> LDS (async engine) ----
    {
        const float* __restrict__ xb = X + col0;
        const int cp = (tid & 15) * 4;            // float offset within row
#pragma unroll
        for (int i = 0; i < 12; ++i) {
            const int row = i * 8 + (tid >> 4);   // 12*8*16 lanes = 1536 chunks
            async_ld_b128(lds_off32(&tile[row * BSTRIDE + cp]),
                          xb + (long long)row * N + cp);
        }
    }
    wait_async0();
    __syncthreads();

    // ---- 24 WMMAs over K=96 ----
    const float* __restrict__ wRow = W + (rowBase + r) * CDIM;
    const float* __restrict__ bp   = &tile[kk * BSTRIDE + wave * 16 + r];

    v8f acc = {};
#pragma unroll
    for (int k0 = 0; k0 < CDIM; k0 += 4) {
        v2f a, b;
        a.x = wRow[k0 + kk];
        a.y = wRow[k0 + kk + 1];
        b.x = bp[k0 * BSTRIDE];                   // ds_load, immediate offset
        b.y = bp[(k0 + 1) * BSTRIDE];
        acc = __builtin_amdgcn_wmma_f32_16x16x4_f32(
                  false, a, false, b, (short)0, acc, false, false);
    }

    const long long col = col0 + wave * 16 + r;
    const int mHalf = hi << 3;
#pragma unroll
    for (int i = 0; i < 8; ++i)
        Y[(long long)(rowBase + mHalf + i) * N + col] = acc[i];
}

// ---------------------------------------------------------------------------
// Kernel 2: depthwise 3x3x3 conv (pad 1) over [288, 64,64,64] plus per-channel
// sum of squares (channels 0..191 = q,k) for the L2 normalization.
// ---------------------------------------------------------------------------
__global__ __launch_bounds__(256) void dwconv3_sumsq(
    const float* __restrict__ in, const float* __restrict__ wdw,
    float* __restrict__ out, float* __restrict__ sumsq)
{
    const int c  = blockIdx.y;
    const int sp = blockIdx.x * 256 + threadIdx.x;     // 0..262143
    const int d  = sp & 63;
    const int w  = (sp >> 6) & 63;
    const int h  = sp >> 12;

    const float* __restrict__ src = in + (long long)c * NTOT;
    const float* __restrict__ wt  = wdw + c * 27;

    float acc = 0.f;
#pragma unroll
    for (int i = 0; i < 3; ++i) {
        const int hh = h + i - 1;
        if ((unsigned)hh < 64u) {
#pragma unroll
            for (int j = 0; j < 3; ++j) {
                const int ww = w + j - 1;
                if ((unsigned)ww < 64u) {
                    const float* __restrict__ row = src + (hh << 12) + (ww << 6);
#pragma unroll
                    for (int kx = 0; kx < 3; ++kx) {
                        const int dd = d + kx - 1;
                        if ((unsigned)dd < 64u)
                            acc += wt[i * 9 + j * 3 + kx] * row[dd];
                    }
                }
            }
        }
    }
    out[(long long)c * NTOT + sp] = acc;

    __shared__ float red[256];
    red[threadIdx.x] = (c < 192) ? acc * acc : 0.f;
    __syncthreads();
    for (int off = 128; off > 0; off >>= 1) {
        if (threadIdx.x < (unsigned)off) red[threadIdx.x] += red[threadIdx.x + off];
        __syncthreads();
    }
    if (threadIdx.x == 0 && c < 192) atomicAdd(&sumsq[c], red[0]);
}

// ---------------------------------------------------------------------------
// Kernel 3: raw channel-attention Gram matrices (16x16xN GEMM per head):
//   attnRaw[h][c][d] += sum_n q[h,c,n] * k[h,d,n]
// One wave per block; 16x128 q/k panels are async-staged into LDS
// (row stride 132 floats -> conflict-free b64 ds reads), 32 WMMAs per panel.
// ---------------------------------------------------------------------------
#define ASTRIDE 132  // 128 data + 4 pad floats

__global__ __launch_bounds__(32) void attn_partial_wmma(
    const float* __restrict__ q, const float* __restrict__ k,
    float* __restrict__ attnRaw, long long N, long long chunk)
{
    __shared__ float tq[16 * ASTRIDE];
    __shared__ float tk[16 * ASTRIDE];

    const int head = blockIdx.y;
    const float* __restrict__ Q = q + (long long)head * 16 * N;
    const float* __restrict__ K = k + (long long)head * 16 * N;

    const int lane = threadIdx.x & 31;
    const int r    = lane & 15;
    const int hi   = lane >> 4;
    const int kk   = hi << 1;

    const long long n0 = (long long)blockIdx.x * chunk;

    const float* __restrict__ ap = &tq[r * ASTRIDE + kk];
    const float* __restrict__ bp = &tk[r * ASTRIDE + kk];

    v8f acc = {};
    for (long long s = 0; s < chunk; s += 128) {
        // stage 16x128 panels of q and k: one async b128 instr per row/matrix
        const long long nb = n0 + s;
#pragma unroll
        for (int row = 0; row < 16; ++row) {
            async_ld_b128(lds_off32(&tq[row * ASTRIDE + lane * 4]),
                          Q + (long long)row * N + nb + lane * 4);
            async_ld_b128(lds_off32(&tk[row * ASTRIDE + lane * 4]),
                          K + (long long)row * N + nb + lane * 4);
        }
        wait_async0();   // single wave: no cross-wave barrier needed

#pragma unroll
        for (int ks = 0; ks < 128; ks += 4) {
            v2f a, b;
            a.x = ap[ks];
            a.y = ap[ks + 1];
            b.x = bp[ks];
            b.y = bp[ks + 1];
            acc = __builtin_amdgcn_wmma_f32_16x16x4_f32(
                      false, a, false, b, (short)0, acc, false, false);
        }
        __builtin_amdgcn_s_barrier();  // wave-local fence before re-staging (no-op cost, single-wave WG)
    }

    const int mHalf = hi << 3;
    float* __restrict__ dst = attnRaw + head * 256;
#pragma unroll
    for (int i = 0; i < 8; ++i)
        atomicAdd(&dst[(mHalf + i) * 16 + r], acc[i]);
}

// ---------------------------------------------------------------------------
// Kernel 4: finalize attention and fold it into the projection:
//   attn = softmax_d( raw[c][d]/(|q_c||k_d|) * temperature[h] )
//   M    = W_proj(96x96) @ blockdiag(attn)  ->  output stage is one GEMM M@v.
// ---------------------------------------------------------------------------
__global__ __launch_bounds__(256) void finalize_attn(
    const float* __restrict__ attnRaw, const float* __restrict__ sumsq,
    const float* __restrict__ w_proj, const float* __restrict__ temperature,
    float* __restrict__ Mmat)
{
    __shared__ float snq[96];
    __shared__ float snk[96];
    __shared__ float sattn[HEADS * 256];

    const int tid = threadIdx.x;
    if (tid < 96) {
        snq[tid] = 1.f / fmaxf(sqrtf(sumsq[tid]),      1e-12f);
        snk[tid] = 1.f / fmaxf(sqrtf(sumsq[96 + tid]), 1e-12f);
    }
    __syncthreads();

    for (int idx = tid; idx < HEADS * 256; idx += 256) {
        const int h = idx >> 8, c = (idx >> 4) & 15, d = idx & 15;
        sattn[idx] = attnRaw[idx] * snq[h * 16 + c] * snk[h * 16 + d] * temperature[h];
    }
    __syncthreads();

    if (tid < 96) {                      // per-row softmax over last dim (16)
        float* row = &sattn[(tid >> 4) * 256 + (tid & 15) * 16];
        float m = row[0];
#pragma unroll
        for (int i = 1; i < 16; ++i) m = fmaxf(m, row[i]);
        float s = 0.f;
#pragma unroll
        for (int i = 0; i < 16; ++i) { const float e = __expf(row[i] - m); row[i] = e; s += e; }
        const float inv = 1.f / s;
#pragma unroll
        for (int i = 0; i < 16; ++i) row[i] *= inv;
    }
    __syncthreads();

    for (int idx = tid; idx < 96 * 96; idx += 256) {
        const int o = idx / 96, g = idx % 96;
        const int h = g >> 4, gg = g & 15;
        float s = 0.f;
#pragma unroll
        for (int c = 0; c < 16; ++c)
            s += w_proj[o * 96 + h * 16 + c] * sattn[h * 256 + c * 16 + gg];
        Mmat[idx] = s;
    }
}

// ---------------------------------------------------------------------------
extern "C" void kernel_launch(void* const* d_in, const int* in_sizes, int n_in,
                              void* d_out, int out_size, void* d_ws, size_t ws_size,
                              hipStream_t stream) {
    const float* x      = (const float*)d_in[0];   // (1,96,64,64,64)
    const float* w_qkv  = (const float*)d_in[1];   // (288,96)
    const float* w_dw   = (const float*)d_in[2];   // (288,1,3,3,3)
    const float* w_proj = (const float*)d_in[3];   // (96,96)
    const float* temp   = (const float*)d_in[4];   // (6,)
    float* out = (float*)d_out;                    // (1,96,64,64,64)

    const long long N = NTOT;

    // workspace layout (floats)
    float* qkv_pre  = (float*)d_ws;                    // 288*N
    float* qkv_post = qkv_pre  + (long long)QKV_C * N; // 288*N
    float* sumsq    = qkv_post + (long long)QKV_C * N; // 192
    float* attnRaw  = sumsq + 192;                     // 6*256
    float* Mmat     = attnRaw + HEADS * 256;           // 96*96

    hipMemsetAsync(sumsq, 0, (192 + HEADS * 256) * sizeof(float), stream);

    // 1) qkv = W_qkv @ x      [288x96] x [96xN]
    {
        dim3 grid((unsigned)(N / 64), QKV_C / 16);
        gemm_wmma_f32<<<grid, 128, 0, stream>>>(w_qkv, x, qkv_pre, N);
    }
    // 2) depthwise 3x3x3 conv + sum-of-squares for q,k channels
    {
        dim3 grid((unsigned)(N / 256), QKV_C);
        dwconv3_sumsq<<<grid, 256, 0, stream>>>(qkv_pre, w_dw, qkv_post, sumsq);
    }
    // 3) raw Gram matrices q.k^T per head
    {
        dim3 grid(256, HEADS);
        attn_partial_wmma<<<grid, 32, 0, stream>>>(
            qkv_post, qkv_post + (long long)CDIM * N, attnRaw, N, N / 256);
    }
    // 4) normalize + softmax + fold into projection
    finalize_attn<<<1, 256, 0, stream>>>(attnRaw, sumsq, w_proj, temp, Mmat);

    // 5) out = M @ v           [96x96] x [96xN]
    {
        dim3 grid((unsigned)(N / 64), CDIM / 16);
        gemm_wmma_f32<<<grid, 128, 0, stream>>>(
            Mmat, qkv_post + 2LL * CDIM * N, out, N);
    }
}